// InvertGHMM_34651796144432
// MI455X (gfx1250) — compile-verified
//
#include <hip/hip_runtime.h>
#include <hip/hip_bf16.h>
#include <math.h>

typedef float v2f __attribute__((ext_vector_type(2)));
typedef float v8f __attribute__((ext_vector_type(8)));

#define T_LEN 128
#define B_SZ  512
#define D_SZ  256
#define L_SZ  64

static __device__ __forceinline__ v8f wmma_f32_4(v2f a, v2f b, v8f c) {
    // D = A(16x4 f32) * B(4x16 f32) + C(16x16 f32)
    return __builtin_amdgcn_wmma_f32_16x16x4_f32(
        /*neg_a=*/false, a, /*neg_b=*/false, b,
        /*c_mod=*/(short)0, c, /*reuse_a=*/false, /*reuse_b=*/false);
}

// ---------------------------------------------------------------------------
// Kernel 0: prep.  expA[j][l] = softmax over row j of transitions.
// miv[l][d] = means[l][d]/var[d]; msq[l] = sum_d means^2/var;
// iv[d] = 1/var[d]; logc = -0.5*D*log(2pi) - 0.5*sum log var.
// ---------------------------------------------------------------------------
__global__ __launch_bounds__(256) void prep_kernel(
    const float* __restrict__ trans, const float* __restrict__ means,
    const float* __restrict__ gvar,
    float* __restrict__ expA, float* __restrict__ miv,
    float* __restrict__ msq, float* __restrict__ iv, float* __restrict__ logc)
{
    int tid = threadIdx.x;
    if (tid < L_SZ) {
        const float* row = trans + tid * L_SZ;
        float mx = -__builtin_inff();
        for (int l = 0; l < L_SZ; ++l) mx = fmaxf(mx, row[l]);
        float s = 0.f;
        for (int l = 0; l < L_SZ; ++l) s += expf(row[l] - mx);
        float lse = mx + logf(s);
        for (int l = 0; l < L_SZ; ++l) expA[tid * L_SZ + l] = expf(row[l] - lse);
    }
    for (int i = tid; i < D_SZ; i += 256) iv[i] = 1.0f / gvar[i];
    for (int i = tid; i < L_SZ * D_SZ; i += 256) {
        int d = i & (D_SZ - 1);
        miv[i] = means[i] / gvar[d];
    }
    if (tid < L_SZ) {
        float s = 0.f;
        const float* mr = means + tid * D_SZ;
        for (int d = 0; d < D_SZ; ++d) { float m = mr[d]; s += m * m / gvar[d]; }
        msq[tid] = s;
    }
    if (tid == 0) {
        float s = 0.f;
        for (int d = 0; d < D_SZ; ++d) s += logf(gvar[d]);
        // log(2*pi) = 1.8378770664093453
        logc[0] = -0.5f * (float)D_SZ * 1.8378770664093453f - 0.5f * s;
    }
}

// ---------------------------------------------------------------------------
// Kernel 1: density GEMM.  dens[tb][l] = logc - 0.5*xsq[tb] - 0.5*msq[l]
//                                        + sum_d X[tb][d]*miv[l][d]
// Block = 4 waves; each wave computes a 16(row) x 64(label) tile via WMMA f32.
// ---------------------------------------------------------------------------
#define MIV_STRIDE 258

__global__ __launch_bounds__(128) void density_kernel(
    const float* __restrict__ X,      // [T*B][D]
    const float* __restrict__ miv_g,  // [L][D]
    const float* __restrict__ msq_g,  // [L]
    const float* __restrict__ iv_g,   // [D]
    const float* __restrict__ logc_g, // [1]
    float* __restrict__ dens)         // [T*B][L]
{
    __shared__ float s_miv[L_SZ * MIV_STRIDE];
    __shared__ float s_iv[D_SZ];
    __shared__ float s_msq[L_SZ];
    __shared__ float s_xsq[4][16];

    int tid = threadIdx.x;
    for (int i = tid; i < L_SZ * D_SZ; i += 128) {
        int l = i >> 8, d = i & (D_SZ - 1);
        s_miv[l * MIV_STRIDE + d] = miv_g[i];
    }
    for (int i = tid; i < D_SZ; i += 128) s_iv[i] = iv_g[i];
    if (tid < L_SZ) s_msq[tid] = msq_g[tid];
    float logc = logc_g[0];
    __syncthreads();

    int wave = tid >> 5, lane = tid & 31;
    int half = lane >> 4, lr = lane & 15;
    int rowbase = blockIdx.x * 64 + wave * 16;
    int row = rowbase + lr;
    const float* xrow = X + (size_t)row * D_SZ;

    // xsq: lane covers half the row, combine across lane-halves
    float xs = 0.f;
    int dbase = half * 128;
    #pragma unroll 8
    for (int d = 0; d < 128; d += 2) {
        v2f xv = *(const v2f*)(xrow + dbase + d);
        xs += xv.x * xv.x * s_iv[dbase + d] + xv.y * xv.y * s_iv[dbase + d + 1];
    }
    xs += __shfl_xor(xs, 16, 32);
    if (half == 0) s_xsq[wave][lr] = xs;

    v8f acc0 = {}, acc1 = {}, acc2 = {}, acc3 = {};
    #pragma unroll 4
    for (int k0 = 0; k0 < D_SZ; k0 += 4) {
        int kk = k0 + 2 * half;
        v2f a  = *(const v2f*)(xrow + kk);
        v2f b0 = *(const v2f*)(&s_miv[(0 * 16 + lr) * MIV_STRIDE + kk]);
        v2f b1 = *(const v2f*)(&s_miv[(1 * 16 + lr) * MIV_STRIDE + kk]);
        v2f b2 = *(const v2f*)(&s_miv[(2 * 16 + lr) * MIV_STRIDE + kk]);
        v2f b3 = *(const v2f*)(&s_miv[(3 * 16 + lr) * MIV_STRIDE + kk]);
        acc0 = wmma_f32_4(a, b0, acc0);
        acc1 = wmma_f32_4(a, b1, acc1);
        acc2 = wmma_f32_4(a, b2, acc2);
        acc3 = wmma_f32_4(a, b3, acc3);
    }

    v8f accs[4] = {acc0, acc1, acc2, acc3};
    #pragma unroll
    for (int nt = 0; nt < 4; ++nt) {
        int col = nt * 16 + lr;
        float mq = s_msq[col];
        #pragma unroll
        for (int i = 0; i < 8; ++i) {
            int m = i + 8 * half;
            float val = accs[nt][i] - 0.5f * s_xsq[wave][m] - 0.5f * mq + logc;
            dens[(size_t)(rowbase + m) * L_SZ + col] = val;
        }
    }
}

// ---------------------------------------------------------------------------
// Kernel 2: scaled forward recurrence.  Each wave owns 16 batch rows for the
// whole t-loop (barrier-free after init).  Per step:
//   mx_b = max_l alpha;  E = exp(alpha - mx);  P = E * expA (WMMA f32);
//   alpha' = mask ? dens + mx + log(P) : alpha.
// ---------------------------------------------------------------------------
#define AL_STRIDE 66

__global__ __launch_bounds__(128) void forward_kernel(
    const float* __restrict__ dens,   // [T*B][L]
    const float* __restrict__ expA_g, // [L][L]
    const float* __restrict__ masks,  // [T][B]
    float* __restrict__ obj)          // [B]
{
    __shared__ float sA[L_SZ * L_SZ];        // expA, row-major [j][l]
    __shared__ float sAl[L_SZ * AL_STRIDE];  // alpha  [b_local][l]
    __shared__ float sE[L_SZ * AL_STRIDE];   // exp(alpha - mx)
    __shared__ float sMx[L_SZ];

    int tid = threadIdx.x;
    int bb = blockIdx.x * 64;

    for (int i = tid; i < L_SZ * L_SZ; i += 128) sA[i] = expA_g[i];
    const float LOGPI = -4.1588830833596715f;   // -log(64)
    for (int i = tid; i < 64 * L_SZ; i += 128) {
        int bl = i >> 6, l = i & 63;
        sAl[bl * AL_STRIDE + l] = dens[(size_t)(bb + bl) * L_SZ + l] + LOGPI;
    }
    __syncthreads();

    int wave = tid >> 5, lane = tid & 31;
    int half = lane >> 4, lr = lane & 15;
    int brow = wave * 16 + lr;                 // local batch row for max/exp
    float* arow = &sAl[brow * AL_STRIDE];
    float* erow = &sE[brow * AL_STRIDE];

    for (int t = 1; t < T_LEN; ++t) {
        // per-row max over labels (halves split l-range, combine via shuffle)
        float mx = -__builtin_inff();
        int lbase = half * 32;
        #pragma unroll 8
        for (int l = 0; l < 32; ++l) mx = fmaxf(mx, arow[lbase + l]);
        mx = fmaxf(mx, __shfl_xor(mx, 16, 32));
        if (half == 0) sMx[brow] = mx;
        #pragma unroll 8
        for (int l = 0; l < 32; ++l) erow[lbase + l] = __expf(arow[lbase + l] - mx);

        // P = E(16x64) * expA(64x64) for this wave's m-tile, K-loop of WMMA
        v8f acc0 = {}, acc1 = {}, acc2 = {}, acc3 = {};
        #pragma unroll 4
        for (int k0 = 0; k0 < L_SZ; k0 += 4) {
            int kk = k0 + 2 * half;
            v2f a = *(const v2f*)(&sE[(wave * 16 + lr) * AL_STRIDE + kk]);
            v2f b0, b1, b2, b3;
            b0.x = sA[kk * L_SZ + 0 * 16 + lr]; b0.y = sA[(kk + 1) * L_SZ + 0 * 16 + lr];
            b1.x = sA[kk * L_SZ + 1 * 16 + lr]; b1.y = sA[(kk + 1) * L_SZ + 1 * 16 + lr];
            b2.x = sA[kk * L_SZ + 2 * 16 + lr]; b2.y = sA[(kk + 1) * L_SZ + 2 * 16 + lr];
            b3.x = sA[kk * L_SZ + 3 * 16 + lr]; b3.y = sA[(kk + 1) * L_SZ + 3 * 16 + lr];
            acc0 = wmma_f32_4(a, b0, acc0);
            acc1 = wmma_f32_4(a, b1, acc1);
            acc2 = wmma_f32_4(a, b2, acc2);
            acc3 = wmma_f32_4(a, b3, acc3);
        }

        v8f accs[4] = {acc0, acc1, acc2, acc3};
        #pragma unroll
        for (int nt = 0; nt < 4; ++nt) {
            int col = nt * 16 + lr;
            #pragma unroll
            for (int i = 0; i < 8; ++i) {
                int m = i + 8 * half;
                int row = wave * 16 + m;
                float P = accs[nt][i];
                float newv = dens[(size_t)(t * B_SZ + bb + row) * L_SZ + col]
                             + sMx[row] + __logf(P);
                float mval = masks[t * B_SZ + bb + row];
                float old = sAl[row * AL_STRIDE + col];
                sAl[row * AL_STRIDE + col] = mval * newv + (1.0f - mval) * old;
            }
        }
    }

    // objective per batch row: logsumexp over labels
    float mx = -__builtin_inff();
    int lbase = half * 32;
    #pragma unroll 8
    for (int l = 0; l < 32; ++l) mx = fmaxf(mx, arow[lbase + l]);
    mx = fmaxf(mx, __shfl_xor(mx, 16, 32));
    float s = 0.f;
    #pragma unroll 8
    for (int l = 0; l < 32; ++l) s += __expf(arow[lbase + l] - mx);
    s += __shfl_xor(s, 16, 32);
    if (half == 0) obj[bb + brow] = mx + __logf(s);
}

// ---------------------------------------------------------------------------
// Kernel 3: deterministic tree reduction of the 512 per-batch objectives.
// ---------------------------------------------------------------------------
__global__ __launch_bounds__(256) void reduce_kernel(
    const float* __restrict__ obj, float* __restrict__ out)
{
    __shared__ float s[256];
    int tid = threadIdx.x;
    s[tid] = obj[tid] + obj[tid + 256];
    __syncthreads();
    for (int st = 128; st > 0; st >>= 1) {
        if (tid < st) s[tid] += s[tid + st];
        __syncthreads();
    }
    if (tid == 0) out[0] = -s[0] / (float)B_SZ;
}

// ---------------------------------------------------------------------------
extern "C" void kernel_launch(void* const* d_in, const int* in_sizes, int n_in,
                              void* d_out, int out_size, void* d_ws, size_t ws_size,
                              hipStream_t stream) {
    (void)in_sizes; (void)n_in; (void)out_size; (void)ws_size;
    // inputs: words(int, unused), pretrained_embed, masks, transitions, means, gaussian_var
    const float* X     = (const float*)d_in[1];  // [T][B][D]
    const float* masks = (const float*)d_in[2];  // [T][B]
    const float* trans = (const float*)d_in[3];  // [L][L]
    const float* means = (const float*)d_in[4];  // [L][D]
    const float* gvar  = (const float*)d_in[5];  // [D]
    float* out = (float*)d_out;

    float* w    = (float*)d_ws;
    float* dens = w;                                 // T*B*L = 4,194,304
    float* expA = dens + (size_t)T_LEN * B_SZ * L_SZ;// 4096
    float* miv  = expA + L_SZ * L_SZ;                // 16384
    float* msq  = miv + L_SZ * D_SZ;                 // 64
    float* iv   = msq + L_SZ;                        // 256
    float* logc = iv + D_SZ;                         // 1 (+pad)
    float* obj  = logc + 16;                         // 512

    prep_kernel<<<1, 256, 0, stream>>>(trans, means, gvar, expA, miv, msq, iv, logc);
    density_kernel<<<(T_LEN * B_SZ) / 64, 128, 0, stream>>>(X, miv, msq, iv, logc, dens);
    forward_kernel<<<B_SZ / 64, 128, 0, stream>>>(dens, expA, masks, obj);
    reduce_kernel<<<1, 256, 0, stream>>>(obj, out);
}